// PixelFilter_32779190403118
// MI455X (gfx1250) — compile-verified
//
#include <hip/hip_runtime.h>

#define RADIUS 3
#define HH 256
#define WW 256
#define HWPLANE (HH * WW)
#define TWPIX 128          // tile width in pixels
#define TH 8               // tile height
#define EPS 1e-9f

// LDS row layout (floats): [0..2]=pad, [3]=left halo, [4..131]=interior,
// [132]=right halo, [133..135]=pad  -> interior starts 16B-aligned.
#define TROWS (TH + 2)     // 10
#define TSTRIDE 136        // row stride (floats); 136*4 B = multiple of 16
#define INT_COL 4          // first interior column
#define NQUAD (TWPIX / 4)  // 32 float4s per interior row
#define NCR (3 * TROWS)    // 30 (channel,row) pairs to stage

typedef float v4f __attribute__((ext_vector_type(4)));

__device__ __forceinline__ int reflect_idx(int v, int n) {
    v = (v < 0) ? -v : v;
    v = (v > n - 1) ? (2 * (n - 1) - v) : v;
    return v;
}

__device__ __forceinline__ unsigned lds_byte_off(const void* p) {
    // LLVM local->flat addrspacecast places the AS(3) byte offset in addr[31:0].
    return (unsigned)(unsigned long long)p;
}

__device__ __forceinline__ void async_copy_b128(unsigned ldsb,
                                                const float* gp) {
    asm volatile("global_load_async_to_lds_b128 %0, %1, off"
                 :: "v"(ldsb), "v"((unsigned long long)gp)
                 : "memory");
}

__global__ void __launch_bounds__(256)
pixel_filter_kernel(const float* __restrict__ x,
                    const float* __restrict__ sigma,
                    float* __restrict__ out) {
    __shared__ __align__(16) float tile[3][TROWS][TSTRIDE];

    const int tid = threadIdx.x;
    const int q   = tid & 31;                                   // lane (divergent)
    // tid>>5 is constant within a wave32 -> force it into an SGPR so all
    // derived addressing becomes scalar (SALU + SADDR-form VMEM).
    const int wv  = __builtin_amdgcn_readfirstlane(tid >> 5);   // wave id 0..7
    const int bx = blockIdx.x * TWPIX;  // tile origin w (0 or 128)
    const int by = blockIdx.y * TH;     // tile origin h
    const int b  = blockIdx.z;          // batch

    const float* xb = x + (size_t)b * 3 * HWPLANE;

    // ---- Stage x halo tile into LDS with gfx1250 async global->LDS copies ----
    // Interior: 30 (c,r) rows x 32 float4 lanes = 960 async b128 ops.
    // (c,r) is wave-uniform each round -> scalar decomposition, scalar
    // reflection, scalar row bases; only base+16*q is per-lane.
#pragma unroll
    for (int rnd = 0; rnd < 4; ++rnd) {
        const int cr = wv + 8 * rnd;            // uniform, 0..31
        if (cr < NCR) {                          // whole-wave predicate
            const int c = cr / TROWS;            // scalar
            const int r = cr % TROWS;            // scalar
            const int gh = reflect_idx(by + r - 1, HH);
            const float* grow = xb + ((size_t)c * HH + gh) * WW + bx;  // scalar
            const unsigned lrow = lds_byte_off(&tile[c][r][INT_COL]);  // scalar
            async_copy_b128(lrow + 16u * q, grow + 4 * q);
        }
    }
    // Halo columns: 3ch * 10 rows * 2 sides = 60 async b32 ops (one round).
    if (tid < NCR * 2) {
        const int c    = tid / (TROWS * 2);
        const int rem  = tid % (TROWS * 2);
        const int r    = rem >> 1;
        const int side = rem & 1;
        const int gh = reflect_idx(by + r - 1, HH);
        const int gw = side ? reflect_idx(bx + TWPIX, WW)
                            : reflect_idx(bx - 1, WW);
        const int lcol = side ? (INT_COL + TWPIX) : (INT_COL - 1);
        const unsigned long long gaddr =
            (unsigned long long)(xb + ((size_t)c * HH + gh) * WW + gw);
        const unsigned ldsb = lds_byte_off(&tile[c][r][lcol]);
        asm volatile("global_load_async_to_lds_b32 %0, %1, off"
                     :: "v"(ldsb), "v"(gaddr)
                     : "memory");
    }
    // Wait this wave's async copies (ASYNCcnt), then sync all waves in the WG.
    asm volatile("s_wait_asynccnt 0" ::: "memory");
    __syncthreads();

    // ---- Per-thread work: 4 consecutive pixels, all 3 channels ----
    const int ty = wv;                  // row within tile (scalar)
    const int h  = by + ty;             // scalar
    const int w0 = bx + 4 * q;          // first of 4 pixels (16B aligned)

    // sigma: 151 MB read exactly once -> non-temporal b128 so it doesn't
    // evict the reusable x halos from the 192 MB L2. Row base is scalar,
    // per-lane offset is 16*q -> SADDR-form global_load_b128.
    const float* srow = sigma + (size_t)b * (RADIUS * RADIUS) * HWPLANE
                              + (size_t)h * WW;
    v4f s[RADIUS * RADIUS];
    float sum0 = 0.f, sum1 = 0.f, sum2 = 0.f, sum3 = 0.f;
#pragma unroll
    for (int k = 0; k < RADIUS * RADIUS; ++k) {
        s[k] = __builtin_nontemporal_load(
                   (const v4f*)(srow + (size_t)k * HWPLANE + w0));
        sum0 += s[k].x; sum1 += s[k].y; sum2 += s[k].z; sum3 += s[k].w;
    }
    const float inv0 = 1.0f / (sum0 + EPS);
    const float inv1 = 1.0f / (sum1 + EPS);
    const float inv2 = 1.0f / (sum2 + EPS);
    const float inv3 = 1.0f / (sum3 + EPS);

    float* orow = out + (size_t)b * 3 * HWPLANE + (size_t)h * WW;
#pragma unroll
    for (int c = 0; c < 3; ++c) {
        float a0 = 0.f, a1 = 0.f, a2 = 0.f, a3 = 0.f;
#pragma unroll
        for (int i = 0; i < RADIUS; ++i) {
            // 6 consecutive x values cover the 3 horizontal taps of 4 pixels.
            const float* row = &tile[c][ty + i][INT_COL - 1 + 4 * q];
            const float v0 = row[0];
            const float v1 = row[1];
            const float v2 = row[2];
            const float v3 = row[3];
            const float v4 = row[4];
            const float v5 = row[5];
            const v4f s0 = s[RADIUS * i + 0];   // tap jj=0 weights, 4 pixels
            const v4f s1 = s[RADIUS * i + 1];   // tap jj=1
            const v4f s2 = s[RADIUS * i + 2];   // tap jj=2
            a0 = fmaf(s0.x, v0, a0); a0 = fmaf(s1.x, v1, a0); a0 = fmaf(s2.x, v2, a0);
            a1 = fmaf(s0.y, v1, a1); a1 = fmaf(s1.y, v2, a1); a1 = fmaf(s2.y, v3, a1);
            a2 = fmaf(s0.z, v2, a2); a2 = fmaf(s1.z, v3, a2); a2 = fmaf(s2.z, v4, a2);
            a3 = fmaf(s0.w, v3, a3); a3 = fmaf(s1.w, v4, a3); a3 = fmaf(s2.w, v5, a3);
        }
        v4f res;
        res.x = a0 * inv0;
        res.y = a1 * inv1;
        res.z = a2 * inv2;
        res.w = a3 * inv3;
        __builtin_nontemporal_store(res,
            (v4f*)(orow + (size_t)c * HWPLANE + w0));
    }
}

extern "C" void kernel_launch(void* const* d_in, const int* in_sizes, int n_in,
                              void* d_out, int out_size, void* d_ws, size_t ws_size,
                              hipStream_t stream) {
    const float* x     = (const float*)d_in[0];
    const float* sigma = (const float*)d_in[1];
    float* out         = (float*)d_out;

    const int batch = in_sizes[0] / (3 * HWPLANE);   // 64
    dim3 grid(WW / TWPIX, HH / TH, batch);           // (2, 32, 64)
    dim3 block(256);
    pixel_filter_kernel<<<grid, block, 0, stream>>>(x, sigma, out);
}